// CNN_88098369175781
// MI455X (gfx1250) — compile-verified
//
#include <hip/hip_runtime.h>
#include <hip/hip_bf16.h>
#include <math.h>

typedef __attribute__((ext_vector_type(2))) float v2f;
typedef __attribute__((ext_vector_type(8))) float v8f;

#define SIGMA   0.005f
#define TILE    256              // time positions per block (8 waves x 2 WMMA tiles x 16)
#define EEG_S   (TILE + 8)       // 264: LDS row stride of the eeg tile
#define WP_S    148              // padded LDS stride for the 16x144 weight tile
#define OFFK(K) (((K) / 9) * EEG_S + ((K) % 9))

// ------------------------------------------------------------------
// Kernel 0: fold projA/projB + diagonal sinc + conv_w[0]/conv_w[2]
// into two effective 10x23 single-channel filters Wc; zero the 30
// running-max accumulators.
// ------------------------------------------------------------------
__global__ void cnn_init_kernel(const float* __restrict__ mu,
                                const float* __restrict__ projA,
                                const float* __restrict__ projB,
                                const float* __restrict__ conv_w,
                                unsigned*    __restrict__ feats,
                                float*       __restrict__ Wc) {
    __shared__ float diag[16][15];
    const int tid = threadIdx.x;
    if (tid < 30) feats[tid] = 0u;          // relu output >= 0 -> uint-max trick valid
    if (tid < 16) {
        const float muc = mu[tid];
        float row[15];
        bool hit = false;
        #pragma unroll
        for (int j = 0; j < 15; ++j) {
            const float kl = (float)(j - 7) / 7.0f;       // linspace(-1,1,15)
            const float kk = (kl - muc) / SIGMA;
            if (fabsf(kk) < 1e-5f) hit = true;
            row[j] = kk;
        }
        #pragma unroll
        for (int j = 0; j < 15; ++j) {
            const float kk = hit ? row[j] - 5e-5f : row[j];
            const float p  = 3.14159265358979323846f * kk;
            diag[tid][j] = sinf(p) / p;
        }
    }
    __syncthreads();
    // Wc[s][o][m] = sum_i proj[i] * sum_k conv_w[s2][o][i][k] * diag[i][m-k]
    for (int idx = tid; idx < 460; idx += blockDim.x) {
        const int s   = idx / 230;            // 0 = stream A, 1 = stream B
        const int rem = idx % 230;
        const int o   = rem / 23;
        const int m   = rem % 23;
        const float* pw = s ? projB : projA;
        const int sbase = (s ? 2 : 0) * 10;
        float sum = 0.f;
        for (int i = 0; i < 16; ++i) {
            float inner = 0.f;
            int klo = m - 14; if (klo < 0) klo = 0;
            const int khi = (m < 8) ? m : 8;
            for (int k = klo; k <= khi; ++k)
                inner += conv_w[(sbase + o) * 144 + i * 9 + k] * diag[i][m - k];
            sum += pw[i] * inner;
        }
        Wc[idx] = sum;
    }
}

// ------------------------------------------------------------------
// Kernel 1: fused conv + relu + global max for all three streams.
// EEG stream: [16x144] x [144 x 32] per wave via f32 WMMA, natural
// K = i*9+k ordering; half-wave K-shift handled by two precomputed
// base pointers so every LDS read uses a pure immediate offset.
// A/B streams: folded 10x23 single-channel conv on VALU.
// ------------------------------------------------------------------
__global__ void __launch_bounds__(256)
cnn_conv_max_kernel(const float* __restrict__ x,
                    const float* __restrict__ conv_w,
                    const float* __restrict__ conv_b,
                    const float* __restrict__ Wc,
                    unsigned*    __restrict__ feats,
                    int T) {
    const int Tout = T - 8;
    const int t0   = blockIdx.x * TILE;
    const int tid  = threadIdx.x;
    const int lane = tid & 31;
    const int wave = tid >> 5;

    __shared__ float    s_eeg[16 * EEG_S];   // [i][j], j = t0..t0+TILE+7
    __shared__ float    s_wavA[TILE + 24];   // t0-7 .. t0+TILE+16
    __shared__ float    s_wavB[TILE + 24];
    __shared__ float    s_Wpad[16 * WP_S];   // [m][K], K = i*9+k, rows 10..15 zero
    __shared__ float    s_WcA[230];
    __shared__ float    s_WcB[230];
    __shared__ float    s_bias[30];
    __shared__ unsigned s_max[30];

    if (tid < 30) { s_max[tid] = 0u; s_bias[tid] = conv_b[tid]; }

    const float* wavA = x;            // row 0  of [18][T]
    const float* eeg  = x + T;        // rows 1..16
    const float* wavB = x + 17 * T;   // row 17

    for (int idx = tid; idx < 16 * EEG_S; idx += 256) {
        const int i = idx / EEG_S, j = idx % EEG_S;
        const int g = t0 + j;
        s_eeg[i * EEG_S + j] = (g < T) ? eeg[i * T + g] : 0.f;
    }
    for (int j = tid; j < TILE + 24; j += 256) {
        const int g = t0 - 7 + j;
        float a = 0.f, b = 0.f;
        if (g >= 0 && g < T) { a = wavA[g]; b = wavB[g]; }
        s_wavA[j] = a; s_wavB[j] = b;
    }
    for (int idx = tid; idx < 16 * 144; idx += 256) {
        const int m = idx / 144, K = idx % 144;
        s_Wpad[m * WP_S + K] = (m < 10) ? conv_w[(10 + m) * 144 + K] : 0.f;
    }
    for (int idx = tid; idx < 230; idx += 256) {
        s_WcA[idx] = Wc[idx];
        s_WcB[idx] = Wc[230 + idx];
    }
    __syncthreads();

    // ---------------- EEG stream via V_WMMA_F32_16X16X4_F32 ----------------
    {
        const int  mn   = lane & 15;         // A row m == B/D column n
        const bool lo   = (lane < 16);       // f32 A/B layout: lanes 16-31 hold K+2
        const int  toff = wave * 32;         // 8 waves x 2 tiles x 16 columns

        const float* Ap     = &s_Wpad[mn * WP_S] + (lo ? 0 : 2);
        const float* Bbase  = &s_eeg[toff + mn];
        const float* Bsame  = Bbase + (lo ? 0 : 2);           // K,K+2 in same eeg row
        const float* Bcross = Bbase + (lo ? 0 : (EEG_S - 7)); // K,K+2 across row edge

        v8f acc0 = {0.f, 0.f, 0.f, 0.f, 0.f, 0.f, 0.f, 0.f};
        v8f acc1 = {0.f, 0.f, 0.f, 0.f, 0.f, 0.f, 0.f, 0.f};
        #pragma unroll
        for (int kb = 0; kb < 36; ++kb) {            // K = 144 = 36 * 4
            const int K0 = 4 * kb;                   // compile-time after unroll
            const int i0 = OFFK(K0);                 // immediate LDS offsets
            const int i1 = OFFK(K0 + 1);
            const float* p0 = ((K0 % 9) > 6)       ? Bcross : Bsame;
            const float* p1 = (((K0 + 1) % 9) > 6) ? Bcross : Bsame;
            v2f a, b0, b1;
            a.x  = Ap[K0];
            a.y  = Ap[K0 + 1];
            b0.x = p0[i0];
            b0.y = p1[i1];
            b1.x = p0[i0 + 16];
            b1.y = p1[i1 + 16];
            acc0 = __builtin_amdgcn_wmma_f32_16x16x4_f32(
                       false, a, false, b0, (short)0, acc0, false, false);
            acc1 = __builtin_amdgcn_wmma_f32_16x16x4_f32(
                       false, a, false, b1, (short)0, acc1, false, false);
        }
        const int mbase = lo ? 0 : 8;                // C/D: VGPR r -> M = r + 8*(lane>=16)
        const int ta    = t0 + toff + mn;
        if (ta < Tout) {
            #pragma unroll
            for (int r = 0; r < 8; ++r) {
                const int m = mbase + r;
                if (m < 10) {
                    const float v = fmaxf(acc0[r] + s_bias[10 + m], 0.f);
                    atomicMax(&s_max[10 + m], __float_as_uint(v));
                }
            }
        }
        if (ta + 16 < Tout) {
            #pragma unroll
            for (int r = 0; r < 8; ++r) {
                const int m = mbase + r;
                if (m < 10) {
                    const float v = fmaxf(acc1[r] + s_bias[10 + m], 0.f);
                    atomicMax(&s_max[10 + m], __float_as_uint(v));
                }
            }
        }
    }

    // ---------------- A / B streams: 10 x 23 conv on VALU ----------------
    {
        const int t = t0 + tid;                      // one position, both streams
        if (t < Tout) {
            float accA[10], accB[10];
            #pragma unroll
            for (int o = 0; o < 10; ++o) { accA[o] = 0.f; accB[o] = 0.f; }
            #pragma unroll
            for (int m = 0; m < 23; ++m) {
                const float xa = s_wavA[tid + m];
                const float xb = s_wavB[tid + m];
                #pragma unroll
                for (int o = 0; o < 10; ++o) {
                    accA[o] += s_WcA[o * 23 + m] * xa;
                    accB[o] += s_WcB[o * 23 + m] * xb;
                }
            }
            #pragma unroll
            for (int o = 0; o < 10; ++o) {
                const float va = fmaxf(accA[o] + s_bias[o], 0.f);
                const float vb = fmaxf(accB[o] + s_bias[20 + o], 0.f);
                atomicMax(&s_max[o],      __float_as_uint(va));
                atomicMax(&s_max[20 + o], __float_as_uint(vb));
            }
        }
    }

    __syncthreads();
    if (tid < 30) atomicMax(&feats[tid], s_max[tid]);
}

// ------------------------------------------------------------------
// Kernel 2: 30 -> sigmoid(30x30) -> sigmoid(2x30)
// ------------------------------------------------------------------
__global__ void cnn_fc_kernel(const unsigned* __restrict__ feats,
                              const float* __restrict__ fc1_w,
                              const float* __restrict__ fc1_b,
                              const float* __restrict__ fc2_w,
                              const float* __restrict__ fc2_b,
                              float* __restrict__ out) {
    __shared__ float f[30], h[30];
    const int tid = threadIdx.x;
    if (tid < 30) f[tid] = __uint_as_float(feats[tid]);
    __syncthreads();
    if (tid < 30) {
        float s = fc1_b[tid];
        #pragma unroll
        for (int j = 0; j < 30; ++j) s += fc1_w[tid * 30 + j] * f[j];
        h[tid] = 1.f / (1.f + expf(-s));
    }
    __syncthreads();
    if (tid < 2) {
        float s = fc2_b[tid];
        #pragma unroll
        for (int j = 0; j < 30; ++j) s += fc2_w[tid * 30 + j] * h[j];
        out[tid] = 1.f / (1.f + expf(-s));
    }
}

extern "C" void kernel_launch(void* const* d_in, const int* in_sizes, int n_in,
                              void* d_out, int out_size, void* d_ws, size_t ws_size,
                              hipStream_t stream) {
    const float* x      = (const float*)d_in[0];
    const float* mu     = (const float*)d_in[1];
    const float* projA  = (const float*)d_in[2];
    const float* projB  = (const float*)d_in[3];
    const float* conv_w = (const float*)d_in[4];
    const float* conv_b = (const float*)d_in[5];
    const float* fc1_w  = (const float*)d_in[6];
    const float* fc1_b  = (const float*)d_in[7];
    const float* fc2_w  = (const float*)d_in[8];
    const float* fc2_b  = (const float*)d_in[9];
    float* out = (float*)d_out;

    const int T = in_sizes[0] / 18;

    unsigned* feats = (unsigned*)d_ws;          // 30 running-max slots (uint bits)
    float*    Wc    = (float*)d_ws + 64;        // 460 floats: WcA[230] ++ WcB[230]

    cnn_init_kernel<<<1, 256, 0, stream>>>(mu, projA, projB, conv_w, feats, Wc);

    const int Tout = T - 8;
    const int nb   = (Tout + TILE - 1) / TILE;
    cnn_conv_max_kernel<<<nb, 256, 0, stream>>>(x, conv_w, conv_b, Wc, feats, T);

    cnn_fc_kernel<<<1, 32, 0, stream>>>(feats, fc1_w, fc1_b, fc2_w, fc2_b, out);
}